// SRenderY_55362128445445
// MI455X (gfx1250) — compile-verified
//
#include <hip/hip_runtime.h>

#define Bc  8
#define Vc  5023
#define Fc  9976
#define Hc  512
#define UVc 256
#define HHc (Hc*Hc)

typedef __attribute__((ext_vector_type(2))) float v2f;
typedef __attribute__((ext_vector_type(8))) float v8f;

// ---------------------------------------------------------------------------
// Kernel 1: zero the normal accumulator
// ---------------------------------------------------------------------------
__global__ __launch_bounds__(256) void zero_kernel(float* __restrict__ p, int n) {
    int i = blockIdx.x * 256 + threadIdx.x;
    if (i < n) p[i] = 0.0f;
}

// ---------------------------------------------------------------------------
// Kernel 2: per-face cross products, scatter-add into per-vertex accumulator
//   n1 = cross(t2-t1, t0-t1) -> vertex faces[f][1]
//   n2 = cross(t0-t2, t1-t2) -> vertex faces[f][2]
//   n0 = cross(t1-t0, t2-t0) -> vertex faces[f][0]
// ---------------------------------------------------------------------------
__global__ __launch_bounds__(256) void face_acc_kernel(const float* __restrict__ verts,
                                                       const int*   __restrict__ faces,
                                                       float*       __restrict__ acc) {
    int t = blockIdx.x * 256 + threadIdx.x;
    if (t >= Bc * Fc) return;
    int b = t / Fc;
    int f = t - b * Fc;
    int i0 = faces[f*3+0], i1 = faces[f*3+1], i2 = faces[f*3+2];
    const float* vb = verts + (size_t)b * Vc * 3;
    float p0x = vb[i0*3+0], p0y = vb[i0*3+1], p0z = vb[i0*3+2];
    float p1x = vb[i1*3+0], p1y = vb[i1*3+1], p1z = vb[i1*3+2];
    float p2x = vb[i2*3+0], p2y = vb[i2*3+1], p2z = vb[i2*3+2];
    float* ab = acc + (size_t)b * Vc * 3;

    // n1 = (p2-p1) x (p0-p1)
    {
        float ax = p2x-p1x, ay = p2y-p1y, az = p2z-p1z;
        float bx = p0x-p1x, by = p0y-p1y, bz = p0z-p1z;
        atomicAdd(&ab[i1*3+0], ay*bz - az*by);
        atomicAdd(&ab[i1*3+1], az*bx - ax*bz);
        atomicAdd(&ab[i1*3+2], ax*by - ay*bx);
    }
    // n2 = (p0-p2) x (p1-p2)
    {
        float ax = p0x-p2x, ay = p0y-p2y, az = p0z-p2z;
        float bx = p1x-p2x, by = p1y-p2y, bz = p1z-p2z;
        atomicAdd(&ab[i2*3+0], ay*bz - az*by);
        atomicAdd(&ab[i2*3+1], az*bx - ax*bz);
        atomicAdd(&ab[i2*3+2], ax*by - ay*bx);
    }
    // n0 = (p1-p0) x (p2-p0)
    {
        float ax = p1x-p0x, ay = p1y-p0y, az = p1z-p0z;
        float bx = p2x-p0x, by = p2y-p0y, bz = p2z-p0z;
        atomicAdd(&ab[i0*3+0], ay*bz - az*by);
        atomicAdd(&ab[i0*3+1], az*bx - ax*bz);
        atomicAdd(&ab[i0*3+2], ax*by - ay*bx);
    }
}

// ---------------------------------------------------------------------------
// Kernel 3: normalize accumulated normals in place
// ---------------------------------------------------------------------------
__global__ __launch_bounds__(256) void normalize_kernel(float* __restrict__ acc) {
    int t = blockIdx.x * 256 + threadIdx.x;
    if (t >= Bc * Vc) return;
    float x = acc[t*3+0], y = acc[t*3+1], z = acc[t*3+2];
    float n = sqrtf(x*x + y*y + z*z);
    float inv = 1.0f / fmaxf(n, 1e-6f);
    acc[t*3+0] = x * inv;
    acc[t*3+1] = y * inv;
    acc[t*3+2] = z * inv;
}

// ---------------------------------------------------------------------------
// Kernel 4: build compact attribute record per (b, face, corner):
//   [gu, gv, nx, ny, nz, pad, pad, pad]   stride 8 floats (32B, float4-aligned)
//   gu = 2u-1,  gv = -(2v-1)  (the grid_sample coordinates)
// ---------------------------------------------------------------------------
__global__ __launch_bounds__(256) void attr_kernel(const float* __restrict__ normals,
                                                   const float* __restrict__ uvc,
                                                   const int*   __restrict__ faces,
                                                   const int*   __restrict__ uvfaces,
                                                   float*       __restrict__ attr) {
    int t = blockIdx.x * 256 + threadIdx.x;           // t = (b*Fc + f)*3 + j
    if (t >= Bc * Fc * 3) return;
    int j  = t % 3;
    int bf = t / 3;
    int b  = bf / Fc;
    int f  = bf - b * Fc;
    int v   = faces[f*3+j];
    int uvv = uvfaces[f*3+j];
    float u  = uvc[uvv*2+0];
    float vv = uvc[uvv*2+1];
    const float* nb = normals + ((size_t)b * Vc + v) * 3;
    float* rec = attr + (size_t)t * 8;
    rec[0] = 2.0f * u - 1.0f;
    rec[1] = -(2.0f * vv - 1.0f);
    rec[2] = nb[0];
    rec[3] = nb[1];
    rec[4] = nb[2];
    rec[5] = 0.0f; rec[6] = 0.0f; rec[7] = 0.0f;
}

// ---------------------------------------------------------------------------
// Kernel 5: fused per-pixel render. One wave (32 lanes) = 16 pixels.
//   shading(3xN) = [lights*SH_CONST](3x9, zero-padded 16x12) x basis(9xN)
//   via 3x V_WMMA_F32_16X16X4_F32 (K chunks 0-3, 4-7, 8-11).
//   A lane layout (16x4 f32): lanes 0-15 {K0,K1}, lanes 16-31 {K2,K3}, M=lane&15
//   B lane layout (4x16 f32): lanes 0-15 {K0,K1}, lanes 16-31 {K2,K3}, N=lane&15
//   D: lanes 0-15, VGPR r -> (channel M=r, pixel N=lane) — matches per-pixel albedo.
// ---------------------------------------------------------------------------
__global__ __launch_bounds__(256) void pixel_kernel(const float* __restrict__ attr,
                                                    const float* __restrict__ albedos,
                                                    const float* __restrict__ lights,
                                                    const float* __restrict__ bary,
                                                    const int*   __restrict__ p2f,
                                                    float*       __restrict__ out) {
    const float SHC0 = 0.2820947918f;
    const float SHC1 = 1.0233267079f;   // = SHC2 = SHC3
    const float SHC4 = 0.8580855308f;   // = SHC5 = SHC6
    const float SHC7 = 0.4290427654f;
    const float SHC8 = 0.2477079561f;

    const int  lane = threadIdx.x & 31;
    const int  sub  = lane & 15;
    const bool hi   = lane >= 16;
    const long wave = (long)blockIdx.x * 8 + (threadIdx.x >> 5);
    const long pix  = wave * 16 + sub;                 // always < Bc*Hc*Hc (exact grid)
    const int  b    = (int)(pix / HHc);
    const int  hw   = (int)(pix - (long)b * HHc);

    const int   fidx   = p2f[pix];
    const bool  masked = fidx < 0;
    const int   idx    = masked ? 0 : fidx;
    const float msk    = masked ? 0.0f : 1.0f;

    const float w0 = bary[pix*3+0];
    const float w1 = bary[pix*3+1];
    const float w2 = bary[pix*3+2];

    // Gather 3 corner records (float4 + scalar each)
    const float* r0 = attr + (size_t)idx * 24;
    const float* r1 = r0 + 8;
    const float* r2 = r0 + 16;
    float4 c0 = *(const float4*)r0; float n0z = r0[4];
    float4 c1 = *(const float4*)r1; float n1z = r1[4];
    float4 c2 = *(const float4*)r2; float n2z = r2[4];

    float gu = msk * (w0*c0.x + w1*c1.x + w2*c2.x);
    float gv = msk * (w0*c0.y + w1*c1.y + w2*c2.y);
    float nx = msk * (w0*c0.z + w1*c1.z + w2*c2.z);
    float ny = msk * (w0*c0.w + w1*c1.w + w2*c2.w);
    float nz = msk * (w0*n0z  + w1*n1z  + w2*n2z );

    // --- bilinear albedo sample (zero padding outside) ---
    float gx = (gu + 1.0f) * (UVc * 0.5f) - 0.5f;
    float gy = (gv + 1.0f) * (UVc * 0.5f) - 0.5f;
    float x0f = floorf(gx), y0f = floorf(gy);
    float wx = gx - x0f, wy = gy - y0f;
    int x0 = (int)x0f, y0 = (int)y0f;
    const float* ab = albedos + (size_t)b * 3 * UVc * UVc;
    float alb0 = 0.f, alb1 = 0.f, alb2 = 0.f;
    #pragma unroll
    for (int tapi = 0; tapi < 4; ++tapi) {
        int ix = x0 + (tapi & 1);
        int iy = y0 + (tapi >> 1);
        float twx = (tapi & 1)  ? wx : (1.0f - wx);
        float twy = (tapi >> 1) ? wy : (1.0f - wy);
        float w = twx * twy;
        float valid = (ix >= 0 && ix < UVc && iy >= 0 && iy < UVc) ? w : 0.0f;
        int ixc = min(max(ix, 0), UVc - 1);
        int iyc = min(max(iy, 0), UVc - 1);
        size_t o = (size_t)iyc * UVc + ixc;
        alb0 += ab[o              ] * valid;
        alb1 += ab[o +   UVc*UVc  ] * valid;
        alb2 += ab[o + 2*UVc*UVc  ] * valid;
    }

    // --- SH basis for this lane's pixel ---
    float bs0 = 1.0f;
    float bs1 = nx, bs2 = ny, bs3 = nz;
    float bs4 = nx*ny, bs5 = nx*nz, bs6 = ny*nz;
    float bs7 = nx*nx - ny*ny;
    float bs8 = 3.0f*nz*nz - 1.0f;

    // --- A matrix values: coeff(c, k) = lights[b, k, c] * SH_CONST[k], rows c>=3 zero
    const int c  = sub;
    const int cc = (c < 3) ? c : 0;
    const float cz = (c < 3) ? 1.0f : 0.0f;
    const float* Lb = lights + b * 27;
    float L0 = Lb[0*3+cc] * SHC0 * cz;
    float L1 = Lb[1*3+cc] * SHC1 * cz;
    float L2 = Lb[2*3+cc] * SHC1 * cz;
    float L3 = Lb[3*3+cc] * SHC1 * cz;
    float L4 = Lb[4*3+cc] * SHC4 * cz;
    float L5 = Lb[5*3+cc] * SHC4 * cz;
    float L6 = Lb[6*3+cc] * SHC4 * cz;
    float L7 = Lb[7*3+cc] * SHC7 * cz;
    float L8 = Lb[8*3+cc] * SHC8 * cz;

    v8f acc = {0.f, 0.f, 0.f, 0.f, 0.f, 0.f, 0.f, 0.f};
    v2f Av, Bv;

    // chunk 0: K = 0..3
    Av.x = hi ? L2  : L0;   Av.y = hi ? L3  : L1;
    Bv.x = hi ? bs2 : bs0;  Bv.y = hi ? bs3 : bs1;
    acc = __builtin_amdgcn_wmma_f32_16x16x4_f32(false, Av, false, Bv, (short)0, acc, false, false);

    // chunk 1: K = 4..7
    Av.x = hi ? L6  : L4;   Av.y = hi ? L7  : L5;
    Bv.x = hi ? bs6 : bs4;  Bv.y = hi ? bs7 : bs5;
    acc = __builtin_amdgcn_wmma_f32_16x16x4_f32(false, Av, false, Bv, (short)0, acc, false, false);

    // chunk 2: K = 8..11 (9..11 zero pad)
    Av.x = hi ? 0.f : L8;   Av.y = 0.f;
    Bv.x = hi ? 0.f : bs8;  Bv.y = 0.f;
    acc = __builtin_amdgcn_wmma_f32_16x16x4_f32(false, Av, false, Bv, (short)0, acc, false, false);

    // lanes 0-15: acc[r] = shading channel r of pixel `sub` (= this lane's pixel)
    if (!hi) {
        float* ob = out + (size_t)b * 3 * HHc + hw;
        ob[0          ] = msk * alb0 * acc[0];
        ob[(size_t)HHc    ] = msk * alb1 * acc[1];
        ob[(size_t)2*HHc  ] = msk * alb2 * acc[2];
    }
}

// ---------------------------------------------------------------------------
extern "C" void kernel_launch(void* const* d_in, const int* in_sizes, int n_in,
                              void* d_out, int out_size, void* d_ws, size_t ws_size,
                              hipStream_t stream) {
    const float* vertices = (const float*)d_in[0];
    // d_in[1] transformed_vertices: dead w.r.t. the output — skipped
    const float* albedos  = (const float*)d_in[2];
    const float* lights   = (const float*)d_in[3];
    const float* uvcoords = (const float*)d_in[4];
    const float* bary     = (const float*)d_in[5];
    const int*   faces    = (const int*)d_in[6];
    const int*   uvfaces  = (const int*)d_in[7];
    const int*   p2f      = (const int*)d_in[8];
    float* out = (float*)d_out;

    float* accN = (float*)d_ws;                        // B*V*3 floats
    float* attr = accN + (size_t)Bc * Vc * 3;          // B*F*3*8 floats (16B aligned)

    const int nAcc = Bc * Vc * 3;
    zero_kernel<<<(nAcc + 255) / 256, 256, 0, stream>>>(accN, nAcc);

    face_acc_kernel<<<(Bc * Fc + 255) / 256, 256, 0, stream>>>(vertices, faces, accN);

    normalize_kernel<<<(Bc * Vc + 255) / 256, 256, 0, stream>>>(accN);

    attr_kernel<<<(Bc * Fc * 3 + 255) / 256, 256, 0, stream>>>(accN, uvcoords, faces, uvfaces, attr);

    const long totalPix = (long)Bc * Hc * Hc;          // 2,097,152
    const long waves    = totalPix / 16;               // 131,072
    const int  blocks   = (int)(waves / 8);            // 16,384 (exact, 8 waves/block)
    pixel_kernel<<<blocks, 256, 0, stream>>>(attr, albedos, lights, bary, p2f, out);
}